// Attention_4810363372509
// MI455X (gfx1250) — compile-verified
//
#include <hip/hip_runtime.h>

// ---------------------------------------------------------------------------
// Fused attention (fwd, out + full p_attn) for MI455X / gfx1250.
// B=2, H=12, S=2048, D=64, fp32. Store-BW bound on the 402MB p_attn write
// (~17us floor @ 23.3 TB/s); compute (25.8 GFLOP) is ~2% of that, so fp32
// WMMA (V_WMMA_F32_16X16X4_F32) keeps reference precision for free.
// Two-pass online-softmax per 128-row query block; QK^T recomputed in pass 2.
// K/V chunks are streamed L2->LDS by the Tensor Data Mover (tensor_load_to_lds,
// TENSORcnt), double-buffered so DMA of chunk i+1 overlaps WMMA of chunk i.
// ---------------------------------------------------------------------------

typedef float v2f  __attribute__((ext_vector_type(2)));
typedef float v8f  __attribute__((ext_vector_type(8)));
typedef unsigned int u32x4 __attribute__((ext_vector_type(4)));
typedef int   i32x4 __attribute__((ext_vector_type(4)));
typedef int   i32x8 __attribute__((ext_vector_type(8)));

#define S_LEN   2048
#define D_DIM   64
#define NHEADS  12
#define QBLOCK  128      // query rows per workgroup (8 waves x 16 rows)
#define KCHUNK  64       // keys per pipelined chunk
#define NCHUNK  (S_LEN / KCHUNK)           // 32
#define NITER   (2 * NCHUNK)               // both passes, one pipelined loop
#define NEG_VAL -1000000000.0f

// LDS layout (floats): Q[128x64] | K double-buf | V double-buf | P scratch
#define LDS_Q   0
#define LDS_K0  8192
#define LDS_K1  12288
#define LDS_V0  16384
#define LDS_V1  20480
#define LDS_P   24576
#define LDS_FLOATS 26624   // 106496 bytes dynamic LDS (<< 320KB per WGP)

__device__ __forceinline__ v8f wmma_f32(v2f a, v2f b, v8f c) {
  // D = A(16x4) * B(4x16) + C(16x16), fp32, wave32.
  return __builtin_amdgcn_wmma_f32_16x16x4_f32(
      /*neg_a=*/false, a, /*neg_b=*/false, b,
      /*c_mod=*/(short)0, c, /*reuse_a=*/false, /*reuse_b=*/false);
}

// TDM 1-D tile load: nelem fp32 elements, global -> LDS (async, TENSORcnt).
// D# per CDNA5 ISA (group0/group1; groups 2-4 unused).
// This toolchain's builtin is the 6-arg form:
//   (u32x4 g0, i32x8 g1, i32x4 g2, i32x4 g3, i32x8 g4, i32 cpol)
__device__ __forceinline__ void tdm_load_1d(unsigned lds_byte_off,
                                            const void* gptr,
                                            unsigned nelem) {
  const unsigned long long ga = (unsigned long long)(uintptr_t)gptr;
  u32x4 g0;
  g0[0] = 1u;                                                 // count=1, user D#
  g0[1] = lds_byte_off;                                       // lds_addr
  g0[2] = (unsigned)ga;                                       // global_addr[31:0]
  g0[3] = ((unsigned)(ga >> 32) & 0x01FFFFFFu) | 0x80000000u; // addr[56:32]|type=2

  i32x8 g1;
  g1[0] = 0x00020000;                                   // data_size=2 (4 bytes)
  g1[1] = (int)(nelem << 16);                           // tensor_dim0[15:0]
  g1[2] = (int)(((nelem >> 16) & 0xFFFFu) | 0x10000u);  // dim0[31:16] | tensor_dim1=1
  g1[3] = (int)(nelem << 16);                           // tile_dim0 = nelem (1-D)
  g1[4] = 0;                                            // tile_dim1/2 unused
  g1[5] = (int)nelem;                                   // tensor_dim0_stride lo
  g1[6] = 0;
  g1[7] = 0;

  i32x4 gz4 = {0, 0, 0, 0};
  i32x8 gz8 = {0, 0, 0, 0, 0, 0, 0, 0};
  __builtin_amdgcn_tensor_load_to_lds(g0, g1, gz4, gz4, gz8, 0);
}

__global__ __launch_bounds__(256) void attn_fused_kernel(
    const float* __restrict__ q, const float* __restrict__ k,
    const float* __restrict__ v, const int* __restrict__ mask,
    float* __restrict__ out, float* __restrict__ pattn)
{
  extern __shared__ float lds[];

  const int tid  = threadIdx.x;
  const int wave = tid >> 5;
  const int lane = tid & 31;
  const int colq = lane & 15;   // col/row-within-16 index per WMMA layout
  const int hi   = lane >> 4;   // selects K-pair (A/B) or M+8 half (C/D)

  const int qb = blockIdx.x;           // 0..15 query blocks
  const int bh = blockIdx.y;           // 0..23 fused batch*head
  const int b  = bh / NHEADS;

  const float* qg   = q + ((size_t)bh * S_LEN + qb * QBLOCK) * D_DIM;
  const float* kg   = k + (size_t)bh * S_LEN * D_DIM;
  const float* vg   = v + (size_t)bh * S_LEN * D_DIM;
  const int*   mrow = mask + (size_t)b * S_LEN;
  const int    qrow0 = qb * QBLOCK + wave * 16;

  // ---- TDM prologue: Q block + chunk-0 K/V pair, issued by wave 0 ----
  if (tid < 32) {
    tdm_load_1d((unsigned)(uintptr_t)(lds + LDS_Q),  qg, QBLOCK * D_DIM);
    tdm_load_1d((unsigned)(uintptr_t)(lds + LDS_K0), kg, KCHUNK * D_DIM);
    tdm_load_1d((unsigned)(uintptr_t)(lds + LDS_V0), vg, KCHUNK * D_DIM);
    __builtin_amdgcn_s_wait_tensorcnt((short)2);   // in-order: Q has landed
  }
  __syncthreads();

  // ---- preload this wave's Q A-fragments (16 rows x 64 K, 16 chunks of K=4)
  // A layout (32-bit 16x4): lane: M=lane&15; VGPR0/1 hold K = (lane>>4)*2 + {0,1}
  v2f qa[16];
  {
    const float* myQ = lds + LDS_Q + (wave * 16 + colq) * D_DIM + hi * 2;
    #pragma unroll
    for (int c = 0; c < 16; ++c) {
      qa[c].x = myQ[c * 4 + 0];
      qa[c].y = myQ[c * 4 + 1];
    }
  }

  // online-softmax state: 8 row-elements per lane (rows r + hi*8), replicated
  // across the 16 lanes of each half-wave.
  float m_run[8], l_run[8];
  #pragma unroll
  for (int r = 0; r < 8; ++r) { m_run[r] = -3.0e38f; l_run[r] = 0.0f; }

  v8f zero8 = {0.f,0.f,0.f,0.f,0.f,0.f,0.f,0.f};
  v8f oacc[4];  // O accumulators: 4 n-tiles of 16 dims
  #pragma unroll
  for (int nt = 0; nt < 4; ++nt) oacc[nt] = zero8;

  float* myP = lds + LDS_P + wave * 256;   // per-wave 16x16 P scratch

  for (int it = 0; it < NITER; ++it) {
    const int pass  = it >> 5;          // 0: running max/sum   1: emit P, O+=PV
    const int kv0   = (it & (NCHUNK - 1)) * KCHUNK;
    const int cur   = it & 1;

    // ---- pipeline: issue DMA for chunk it+1, wait for chunk it ----
    if (tid < 32) {
      if (it + 1 < NITER) {
        const int nkv0 = ((it + 1) & (NCHUNK - 1)) * KCHUNK;
        const int nb   = (it + 1) & 1;
        tdm_load_1d((unsigned)(uintptr_t)(lds + (nb ? LDS_K1 : LDS_K0)),
                    kg + (size_t)nkv0 * D_DIM, KCHUNK * D_DIM);
        tdm_load_1d((unsigned)(uintptr_t)(lds + (nb ? LDS_V1 : LDS_V0)),
                    vg + (size_t)nkv0 * D_DIM, KCHUNK * D_DIM);
        __builtin_amdgcn_s_wait_tensorcnt((short)2);  // current pair complete
      } else {
        __builtin_amdgcn_s_wait_tensorcnt((short)0);
      }
    }
    __syncthreads();   // chunk `it` visible in LDS to all waves

    const float* sK = lds + (cur ? LDS_K1 : LDS_K0);
    const float* sV = lds + (cur ? LDS_V1 : LDS_V0);

    #pragma unroll
    for (int ti = 0; ti < 4; ++ti) {        // 16-key WMMA tiles in chunk
      const int colg = kv0 + ti * 16 + colq;  // global key index for this lane

      // ---- S tile = Q(16x64) * K^T(64x16): 16 chained fp32 WMMAs ----
      v8f acc = zero8;
      const float* myK = sK + (ti * 16 + colq) * D_DIM + hi * 2;
      #pragma unroll
      for (int c = 0; c < 16; ++c) {
        v2f bfr; bfr.x = myK[c * 4 + 0]; bfr.y = myK[c * 4 + 1];
        acc = wmma_f32(qa[c], bfr, acc);
      }

      const bool dead = (mrow[colg] == 0);
      float s[8];
      #pragma unroll
      for (int r = 0; r < 8; ++r)
        s[r] = dead ? NEG_VAL : acc[r] * 0.125f;   // 1/sqrt(64)

      if (pass == 0) {
        // ---- online max/sum update (reduce over the 16-lane half-wave) ----
        #pragma unroll
        for (int r = 0; r < 8; ++r) {
          float tmax = s[r];
          #pragma unroll
          for (int mm = 8; mm >= 1; mm >>= 1)
            tmax = fmaxf(tmax, __shfl_xor(tmax, mm));
          const float mnew = fmaxf(m_run[r], tmax);
          float tsum = __expf(s[r] - mnew);
          #pragma unroll
          for (int mm = 8; mm >= 1; mm >>= 1)
            tsum += __shfl_xor(tsum, mm);
          l_run[r] = l_run[r] * __expf(m_run[r] - mnew) + tsum;
          m_run[r] = mnew;
        }
      } else {
        // ---- normalize, emit p_attn, accumulate O += P * V ----
        #pragma unroll
        for (int r = 0; r < 8; ++r) {
          const float p = __expf(s[r] - m_run[r]) / l_run[r];
          pattn[((size_t)bh * S_LEN + qrow0 + r + hi * 8) * (size_t)S_LEN + colg] = p;
          myP[(r + hi * 8) * 16 + colq] = p;   // C/D layout -> LDS (row-major)
        }
        // reload P in A-fragment layout (intra-wave, DScnt-ordered)
        v2f pa[4];
        const float* myPr = myP + colq * 16 + hi * 2;
        #pragma unroll
        for (int c = 0; c < 4; ++c) { pa[c].x = myPr[c * 4]; pa[c].y = myPr[c * 4 + 1]; }

        // B-fragments of V: B[k][n] = V[key k][dim n]
        const float* vbase = sV + (ti * 16 + hi * 2) * D_DIM + colq;
        #pragma unroll
        for (int c = 0; c < 4; ++c) {
          #pragma unroll
          for (int nt = 0; nt < 4; ++nt) {
            v2f bfr;
            bfr.x = vbase[(c * 4 + 0) * D_DIM + nt * 16];
            bfr.y = vbase[(c * 4 + 1) * D_DIM + nt * 16];
            oacc[nt] = wmma_f32(pa[c], bfr, oacc[nt]);
          }
        }
      }
    }
    __syncthreads();   // all waves done with this buffer before DMA reuses it
  }

  // ---- write O (16 rows x 64 dims per wave) ----
  #pragma unroll
  for (int nt = 0; nt < 4; ++nt)
    #pragma unroll
    for (int r = 0; r < 8; ++r)
      out[((size_t)bh * S_LEN + qrow0 + r + hi * 8) * D_DIM + nt * 16 + colq] = oacc[nt][r];
}

extern "C" void kernel_launch(void* const* d_in, const int* in_sizes, int n_in,
                              void* d_out, int out_size, void* d_ws, size_t ws_size,
                              hipStream_t stream) {
  const float* q    = (const float*)d_in[0];
  const float* k    = (const float*)d_in[1];
  const float* v    = (const float*)d_in[2];
  const int*   mask = (const int*)d_in[3];

  float* out   = (float*)d_out;                         // [2,12,2048,64]
  float* pattn = out + (size_t)2 * 12 * 2048 * 64;      // [2,12,2048,2048]

  dim3 grid(S_LEN / QBLOCK, 2 * NHEADS);                // (16, 24)
  attn_fused_kernel<<<grid, 256, LDS_FLOATS * sizeof(float), stream>>>(
      q, k, v, mask, out, pattn);
}